// Prompt_block_62904091017899
// MI455X (gfx1250) — compile-verified
//
#include <hip/hip_runtime.h>
#include <hip/hip_bf16.h>

#define B_   16
#define L0_  1024
#define L1_  3136
#define C_   768
#define HC_  8
#define LTOT (L0_ + L1_)

typedef float v2f __attribute__((ext_vector_type(2)));
typedef float v8f __attribute__((ext_vector_type(8)));
typedef unsigned int v4u __attribute__((ext_vector_type(4)));
typedef int v8i __attribute__((ext_vector_type(8)));
typedef int v4i __attribute__((ext_vector_type(4)));

// ---------------------------------------------------------------------------
// TDM: DMA one 16-row x 32-col fp32 tile (row stride 768 floats) from global
// memory into LDS with a 34-float LDS row pitch (pad 2 DWORDs every 32 DWORDs,
// via D#.pad_interval=4 (32 dw), D#.pad_amount=1 (2 dw)).  Tracked on
// TENSORcnt; one instruction per wave per tile.
// ---------------------------------------------------------------------------
static __device__ __forceinline__ void tdm_load_tile16x32(unsigned int lds_off,
                                                          const float* gptr) {
    const unsigned long long ga = (unsigned long long)(size_t)gptr;
    v4u g0;
    g0.x = 1u;                                   // count=1 (valid), user mode
    g0.y = lds_off;                              // lds_addr [63:32]
    g0.z = (unsigned int)ga;                     // global_addr [95:64]
    g0.w = (unsigned int)((ga >> 32) & 0x01FFFFFFu) | (2u << 30); // addr hi + type=2
    v8i g1;
    g1[0] = (2 << 16)      // data_size = 4 bytes
          | (1 << 20)      // pad_enable
          | (4 << 22)      // pad_interval: 32 DWORDs
          | (1 << 25);     // pad_amount: 2 DWORDs
    g1[1] = (int)(768u << 16);   // tensor_dim0[15:0] = 768 (bits 79:48, lo half)
    g1[2] = (int)(16u  << 16);   // tensor_dim1[15:0] = 16  (bits 111:80, lo half)
    g1[3] = (int)(32u  << 16);   // tile_dim0 = 32    (bits 127:112)
    g1[4] = 16;                  // tile_dim1 = 16    (bits 143:128); tile_dim2=0
    g1[5] = 768;                 // tensor_dim0_stride[31:0] = 768
    g1[6] = 0;                   // stride hi + tensor_dim1_stride lo
    g1[7] = 0;
    v4i gz4 = {0, 0, 0, 0};      // groups 2/3 unused (2D tensor)
    v8i gz8 = {0, 0, 0, 0, 0, 0, 0, 0};
    __builtin_amdgcn_tensor_load_to_lds(g0, g1, gz4, gz4, gz8, 0);
}

// ---------------------------------------------------------------------------
// Kernel 1: projection  x @ {W00|W01} + bias  ->  x0t / x1t
// One wave computes a 16-row x 8-col tile via v_wmma_f32_16x16x4_f32.
// A-operand tiles are DMA'd by the Tensor Data Mover into per-wave
// double-buffered LDS (no block barriers in the k-loop, TENSORcnt only).
// Block = 4 waves = 64 rows; 64 | 1024 and 64 | 4160, so a block never
// straddles the x0/x1 region boundary -> one weight matrix per block.
// ---------------------------------------------------------------------------
#define NCH (C_ / 32)   // 24 k-chunks of 32
__global__ __launch_bounds__(128) void proj_kernel(
    const float* __restrict__ x,
    const float* __restrict__ W00, const float* __restrict__ b00,
    const float* __restrict__ W01, const float* __restrict__ b01,
    float* __restrict__ x0t, float* __restrict__ x1t)
{
    __shared__ float sW[C_ * HC_];        // W as-is; lanes n>=8 mask loaded value
    __shared__ float sBias[16];
    __shared__ float sA[4][2][16][34];    // per-wave double buffer, pitch 34

    const int tid    = threadIdx.x;
    const int blk    = blockIdx.x;
    const int rowBlk = blk * 64;                 // global row base (b*LTOT + l)
    const int b      = rowBlk / LTOT;
    const int l      = rowBlk - b * LTOT;
    const bool reg0  = (l < L0_);

    const float* W    = reg0 ? W00 : W01;
    const float* bias = reg0 ? b00 : b01;
    float* dst = reg0 ? (x0t + (size_t)(b * L0_ + l) * HC_)
                      : (x1t + (size_t)(b * L1_ + (l - L0_)) * HC_);

    for (int i = tid; i < C_ * HC_; i += 128) sW[i] = W[i];
    if (tid < 16) sBias[tid] = (tid < HC_) ? bias[tid] : 0.0f;

    const int w    = tid >> 5;        // wave in block
    const int lane = tid & 31;
    const int half = lane >> 4;       // K half-split per ISA A/B fragment layout
    const int mn   = lane & 15;       // M for A-frag, N for B-frag
    const size_t xbase = (size_t)(rowBlk + w * 16) * C_;
    const float* gtile = x + xbase;

    const unsigned int ldsA0 = (unsigned int)(size_t)(&sA[w][0][0][0]);
    const unsigned int ldsA1 = (unsigned int)(size_t)(&sA[w][1][0][0]);

    const int  bcol   = mn & 7;        // B column folded to valid range
    const bool bvalid = (mn < HC_);    // lanes n>=8 contribute zero columns

    __syncthreads();   // sW ready (TDM only touches sA)

    v8f acc = {};
    // consume one 16x32 chunk: 8 x v_wmma_f32_16x16x4_f32
    auto consume = [&](int kc, int buf) {
        const int k0 = kc * 32;
        #pragma unroll
        for (int s = 0; s < 8; ++s) {
            const int k = s * 4 + 2 * half;   // per-lane K offset within chunk
            v2f a, bb;
            a.x = sA[w][buf][mn][k];          // addrspace(3) -> ds_load, imm offsets
            a.y = sA[w][buf][mn][k + 1];
            const float b0 = sW[(k0 + k) * HC_ + bcol];
            const float b1 = sW[(k0 + k + 1) * HC_ + bcol];
            bb.x = bvalid ? b0 : 0.0f;        // value select, EXEC untouched
            bb.y = bvalid ? b1 : 0.0f;
            acc = __builtin_amdgcn_wmma_f32_16x16x4_f32(
                      false, a, false, bb, (short)0, acc, false, false);
        }
    };

    tdm_load_tile16x32(ldsA0, gtile);                       // chunk 0 -> buf 0
    for (int kc = 0; kc < NCH - 1; ++kc) {
        const int nbuf = (kc + 1) & 1;
        tdm_load_tile16x32(nbuf ? ldsA1 : ldsA0, gtile + (kc + 1) * 32);
        __builtin_amdgcn_s_wait_tensorcnt(1);               // chunk kc landed
        asm volatile("" ::: "memory");
        consume(kc, kc & 1);
    }
    __builtin_amdgcn_s_wait_tensorcnt(0);
    asm volatile("" ::: "memory");
    consume(NCH - 1, (NCH - 1) & 1);

    // C/D layout: VGPR r -> rows (r, r+8) for lane halves; lane%16 = N
    if (mn < HC_) {
        const float bv = sBias[mn];
        #pragma unroll
        for (int r = 0; r < 8; ++r) {
            const int rowl = w * 16 + r + 8 * half;
            dst[(size_t)rowl * HC_ + mn] = acc[r] + bv;
        }
    }
}

// ---------------------------------------------------------------------------
// Kernel 2: attention pass A -- x0_1[b,l,:] = softmax_m(x0t . x1t^T) @ x1t
// Flash-style online softmax, keys chunked through LDS (broadcast reads).
// ---------------------------------------------------------------------------
#define CHA 784   // 3136 = 4 * 784 ; 784*8*4B = 25 KB LDS
__global__ __launch_bounds__(128) void attnA_kernel(
    const float* __restrict__ x0t, const float* __restrict__ x1t,
    float* __restrict__ x01)
{
    __shared__ float sK[CHA * HC_];
    const int b = blockIdx.x >> 3;
    const int l = (blockIdx.x & 7) * 128 + threadIdx.x;

    float q[HC_];
    const float* qp = x0t + ((size_t)b * L0_ + l) * HC_;
    #pragma unroll
    for (int d = 0; d < HC_; ++d) q[d] = qp[d];

    float mrun = -1e30f, srun = 0.0f;
    float acc[HC_] = {};
    const float* Kb = x1t + (size_t)b * L1_ * HC_;

    for (int ch = 0; ch < L1_ / CHA; ++ch) {
        __syncthreads();
        for (int i = threadIdx.x; i < CHA * HC_; i += 128)
            sK[i] = Kb[(size_t)ch * CHA * HC_ + i];
        if (ch + 1 < L1_ / CHA)
            __builtin_prefetch(&Kb[(size_t)(ch + 1) * CHA * HC_ + threadIdx.x], 0, 1);
        __syncthreads();
        for (int j = 0; j < CHA; ++j) {
            float dot = 0.0f;
            #pragma unroll
            for (int d = 0; d < HC_; ++d) dot += q[d] * sK[j * HC_ + d];
            const float nm   = fmaxf(mrun, dot);
            const float corr = __expf(mrun - nm);
            const float p    = __expf(dot - nm);
            srun = srun * corr + p;
            #pragma unroll
            for (int d = 0; d < HC_; ++d)
                acc[d] = acc[d] * corr + p * sK[j * HC_ + d];
            mrun = nm;
        }
    }
    const float inv = 1.0f / srun;
    float* op = x01 + ((size_t)b * L0_ + l) * HC_;
    #pragma unroll
    for (int d = 0; d < HC_; ++d) op[d] = acc[d] * inv;
}

// ---------------------------------------------------------------------------
// Kernel 3: attention pass B -- x1_0[b,m,:] = softmax_l(x1t . x0t^T) @ x0t
// All 1024 keys (32 KB) fit in LDS in one chunk.
// ---------------------------------------------------------------------------
__global__ __launch_bounds__(256) void attnB_kernel(
    const float* __restrict__ x1t, const float* __restrict__ x0t,
    float* __restrict__ x10)
{
    __shared__ float sK[L0_ * HC_];   // 32 KB
    const int b = blockIdx.y;
    const int l = blockIdx.x * 256 + threadIdx.x;

    const float* Kb = x0t + (size_t)b * L0_ * HC_;
    for (int i = threadIdx.x; i < L0_ * HC_; i += 256) sK[i] = Kb[i];
    __syncthreads();
    if (l >= L1_) return;

    float q[HC_];
    const float* qp = x1t + ((size_t)b * L1_ + l) * HC_;
    #pragma unroll
    for (int d = 0; d < HC_; ++d) q[d] = qp[d];

    float mrun = -1e30f, srun = 0.0f;
    float acc[HC_] = {};
    for (int j = 0; j < L0_; ++j) {
        float dot = 0.0f;
        #pragma unroll
        for (int d = 0; d < HC_; ++d) dot += q[d] * sK[j * HC_ + d];
        const float nm   = fmaxf(mrun, dot);
        const float corr = __expf(mrun - nm);
        const float p    = __expf(dot - nm);
        srun = srun * corr + p;
        #pragma unroll
        for (int d = 0; d < HC_; ++d)
            acc[d] = acc[d] * corr + p * sK[j * HC_ + d];
        mrun = nm;
    }
    const float inv = 1.0f / srun;
    float* op = x10 + ((size_t)b * L1_ + l) * HC_;
    #pragma unroll
    for (int d = 0; d < HC_; ++d) op[d] = acc[d] * inv;
}

// ---------------------------------------------------------------------------
// Kernel 4: fovea + residual:  out[b,l,d] = softmax_l(t)[b,l,d]*t[b,l,d] + add
// One block per (b,d); two-pass block reduction (max, sumexp) over L.
// ---------------------------------------------------------------------------
template <int LEN>
__global__ __launch_bounds__(256) void fovea_kernel(
    const float* __restrict__ t, const float* __restrict__ addv,
    float* __restrict__ outv)
{
    __shared__ float red[256];
    const int b = blockIdx.x >> 3;
    const int d = blockIdx.x & 7;
    const float* tb = t + (size_t)b * LEN * HC_ + d;

    float mx = -1e30f;
    for (int l = threadIdx.x; l < LEN; l += 256)
        mx = fmaxf(mx, tb[(size_t)l * HC_]);
    red[threadIdx.x] = mx; __syncthreads();
    for (int s = 128; s > 0; s >>= 1) {
        if (threadIdx.x < s) red[threadIdx.x] = fmaxf(red[threadIdx.x], red[threadIdx.x + s]);
        __syncthreads();
    }
    mx = red[0]; __syncthreads();

    float sm = 0.0f;
    for (int l = threadIdx.x; l < LEN; l += 256)
        sm += __expf(tb[(size_t)l * HC_] - mx);
    red[threadIdx.x] = sm; __syncthreads();
    for (int s = 128; s > 0; s >>= 1) {
        if (threadIdx.x < s) red[threadIdx.x] += red[threadIdx.x + s];
        __syncthreads();
    }
    const float inv = 1.0f / red[0];

    const float* ab = addv + (size_t)b * LEN * HC_ + d;
    float* ob = outv + (size_t)b * LEN * HC_ + d;
    for (int l = threadIdx.x; l < LEN; l += 256) {
        const float v = tb[(size_t)l * HC_];
        ob[(size_t)l * HC_] = __expf(v - mx) * inv * v + ab[(size_t)l * HC_];
    }
}

// ---------------------------------------------------------------------------
// Kernel 5: output GEMM + transpose:  f[b,c,l] = sum_d xo[b,l,d]*W[d,c] + bias[c]
// Thread-per-l, 192-wide c slice from LDS (broadcast); stores coalesced in l.
// ---------------------------------------------------------------------------
template <int LEN>
__global__ __launch_bounds__(256) void out_kernel(
    const float* __restrict__ xo, const float* __restrict__ W,
    const float* __restrict__ bias, float* __restrict__ f)
{
    __shared__ float sW[HC_][192];
    __shared__ float sB[192];
    const int b  = blockIdx.z;
    const int c0 = blockIdx.y * 192;
    const int l  = blockIdx.x * 256 + threadIdx.x;

    for (int i = threadIdx.x; i < HC_ * 192; i += 256) {
        const int d = i / 192, c = i - d * 192;
        sW[d][c] = W[d * C_ + c0 + c];
    }
    if (threadIdx.x < 192) sB[threadIdx.x] = bias[c0 + threadIdx.x];
    __syncthreads();
    if (l >= LEN) return;

    float xr[HC_];
    const float* xp = xo + ((size_t)b * LEN + l) * HC_;
    #pragma unroll
    for (int d = 0; d < HC_; ++d) xr[d] = xp[d];

    float* fb = f + (size_t)b * C_ * LEN + l;
    for (int c = 0; c < 192; ++c) {
        float v = sB[c];
        #pragma unroll
        for (int d = 0; d < HC_; ++d) v += xr[d] * sW[d][c];
        fb[(size_t)(c0 + c) * LEN] = v;
    }
}

// ---------------------------------------------------------------------------
extern "C" void kernel_launch(void* const* d_in, const int* in_sizes, int n_in,
                              void* d_out, int out_size, void* d_ws, size_t ws_size,
                              hipStream_t stream) {
    const float* x   = (const float*)d_in[0];
    const float* W00 = (const float*)d_in[1];
    const float* b00 = (const float*)d_in[2];
    const float* W01 = (const float*)d_in[3];
    const float* b01 = (const float*)d_in[4];
    const float* W1  = (const float*)d_in[5];
    const float* b1  = (const float*)d_in[6];
    const float* W2  = (const float*)d_in[7];
    const float* b2  = (const float*)d_in[8];

    float* f0 = (float*)d_out;
    float* f1 = f0 + (size_t)B_ * C_ * L0_;

    float* ws  = (float*)d_ws;
    float* x0t = ws;                                 // B*L0*HC
    float* x1t = x0t + (size_t)B_ * L0_ * HC_;       // B*L1*HC
    float* x01 = x1t + (size_t)B_ * L1_ * HC_;       // B*L0*HC (x0_1, then x0o)
    float* x10 = x01 + (size_t)B_ * L0_ * HC_;       // B*L1*HC (x1_0, then x1o)

    // 1) projections (reads the 204 MB x exactly once: TDM -> LDS -> WMMA)
    proj_kernel<<<(B_ * LTOT) / 64, 128, 0, stream>>>(x, W00, b00, W01, b01, x0t, x1t);
    // 2) cross attention, both directions, flash-style (no S materialization)
    attnA_kernel<<<B_ * 8, 128, 0, stream>>>(x0t, x1t, x01);
    attnB_kernel<<<dim3((L1_ + 255) / 256, B_), 256, 0, stream>>>(x1t, x0t, x10);
    // 3) fovea (+ residual), in place: x01 -> x0o, x10 -> x1o
    fovea_kernel<L0_><<<B_ * HC_, 256, 0, stream>>>(x0t, x01, x01);
    fovea_kernel<L1_><<<B_ * HC_, 256, 0, stream>>>(x10, x1t, x10);
    // 4) output GEMMs + transpose to (B, C, H, W) layout (writes 204 MB once)
    out_kernel<L0_><<<dim3((L0_ + 255) / 256, 4, B_), 256, 0, stream>>>(x01, W1, b1, f0);
    out_kernel<L1_><<<dim3((L1_ + 255) / 256, 4, B_), 256, 0, stream>>>(x10, W2, b2, f1);
}